// glocapBlockEM_42631845380379
// MI455X (gfx1250) — compile-verified
//
#include <hip/hip_runtime.h>
#include <math.h>

// Problem constants (from reference setup_inputs)
#define NN 32       // batch
#define CCH 32      // input capsule channels
#define DD 16       // pose dim = P*P
#define SS 196      // W*H = 14*14
#define BB (CCH*SS) // 6272 votes per (n)
#define MM 10       // output capsules
#define PSZ 16      // pose size
#define LAM_INIT 0.001f
#define LAM_ROUT 0.001f
#define EPSF 1e-6f
#define LN2PI 1.8378770664093453f

typedef float v2f __attribute__((ext_vector_type(2)));
typedef float v8f __attribute__((ext_vector_type(8)));

// ---------------------------------------------------------------------------
// Kernel 1: votes V[n][b][m][d] = einsum('ncspq,cmqr->ncsmpr') via
// V_WMMA_F32_16X16X4_F32.  Per channel c this is a GEMM:
//   rows = (n,s,p)  [25088],  K = q [4],  cols = (m,r) [40 -> 3 tiles of 16]
// One wave owns one 16-row tile and all 3 column tiles: the A operand is
// loaded once and reused by 3 single-shot WMMAs (K==4 == P, no K loop).
// ---------------------------------------------------------------------------
__global__ __launch_bounds__(256) void k_votes_wmma(
    const float* __restrict__ l,      // [N][C][D][S]
    const float* __restrict__ w,      // [C][M][4][4]
    float* __restrict__ V)            // [N][B][M][16]
{
  const int lane = threadIdx.x & 31;
  int wid = (int)((blockIdx.x * 256u + threadIdx.x) >> 5);
  const int st = wid % 49; wid /= 49;     // row tile within n (784 rows / 16)
  const int n  = wid % NN;
  const int c  = wid / NN;                // 0..31

  const int half = lane >> 4;             // 0: K=0,1  1: K=2,3
  const int l16  = lane & 15;
  const int k0   = half * 2;

  // A operand (loaded once, reused for all 3 col tiles):
  // row = st*16 + l16 -> (s = row>>2, p = row&3); lp[s,p,q] = l[4p+q][s]
  const int row = st * 16 + l16;
  const int s_  = row >> 2;
  const int p_  = row & 3;
  const float* lb = l + ((size_t)n * CCH + c) * DD * SS;
  v2f A;
  A.x = lb[(size_t)(p_ * 4 + k0)     * SS + s_];
  A.y = lb[(size_t)(p_ * 4 + k0 + 1) * SS + s_];

#pragma unroll
  for (int ct = 0; ct < 3; ++ct) {
    // B operand: col = ct*16 + l16 -> (m = col>>2, r = col&3); w[c][m][q][r]
    const int col = ct * 16 + l16;
    const bool cv = (col < MM * 4);
    const int m_  = cv ? (col >> 2) : 0;
    const int r_  = col & 3;
    const float* wb = w + (((size_t)c * MM + m_) * 4 + k0) * 4 + r_;
    float b0 = wb[0], b1 = wb[4];
    v2f B;
    B.x = cv ? b0 : 0.0f;
    B.y = cv ? b1 : 0.0f;

    v8f C = {};
    // D = A x B + 0   (single WMMA per tile, K=4)
    C = __builtin_amdgcn_wmma_f32_16x16x4_f32(false, A, false, B, (short)0, C,
                                              false, false);

    if (cv) {
      const int mm = col >> 2;
#pragma unroll
      for (int j = 0; j < 8; ++j) {
        const int rr = st * 16 + j + half * 8;   // D row: M = j + 8*(lane/16)
        const int ss2 = rr >> 2, pp = rr & 3;
        const int b = c * SS + ss2;
        V[(((size_t)n * BB + b) * MM + mm) * PSZ + pp * 4 + r_] = C[j];
      }
    }
  }
}

// ---------------------------------------------------------------------------
// zero scratch
// ---------------------------------------------------------------------------
__global__ void k_zero(float* __restrict__ p, int n) {
  int i = blockIdx.x * blockDim.x + threadIdx.x;
  if (i < n) p[i] = 0.0f;
}

// ---------------------------------------------------------------------------
// Kernel 2: SigmaV / SigmaV^2 over b, per (n,m,d).  Covers BOTH the init
// activation (Sum (V-g)^2 = SV2 - 2g SV + B g^2) and the iteration-0 M-step
// (coeff is uniform over b in iter 0).
// grid = N*8 blocks, 256 thr: lane d = t&15, b-group = t>>4, 49 b's each.
// ---------------------------------------------------------------------------
__global__ __launch_bounds__(256) void k_init_acc(
    const float* __restrict__ V, float* __restrict__ S1, float* __restrict__ S2)
{
  const int n = blockIdx.x >> 3;
  const int chunk = blockIdx.x & 7;
  const int d  = threadIdx.x & 15;
  const int bg = threadIdx.x >> 4;
  float s1[MM] = {}, s2[MM] = {};
  for (int i = 0; i < 49; ++i) {
    const int b = chunk * 784 + bg + i * 16;
    const float* vr = V + ((size_t)n * BB + b) * MM * PSZ + d;
#pragma unroll
    for (int m = 0; m < MM; ++m) {
      float v = vr[m * PSZ];
      s1[m] += v;
      s2[m] += v * v;
    }
  }
#pragma unroll
  for (int m = 0; m < MM; ++m) {
    atomicAdd(&S1[(n * MM + m) * PSZ + d], s1[m]);
    atomicAdd(&S2[(n * MM + m) * PSZ + d], s2[m]);
  }
}

// ---------------------------------------------------------------------------
// Kernel 3: finalize init-activation + EM iteration 0 (uniform r).
// One thread per (n,m), loops d.
// ---------------------------------------------------------------------------
__global__ void k_finalize0(
    const float* __restrict__ S1, const float* __restrict__ S2,
    const float* __restrict__ g, const float* __restrict__ beta_a,
    const float* __restrict__ beta_u,
    float* __restrict__ MU, float* __restrict__ SG, float* __restrict__ AO)
{
  int t = blockIdx.x * blockDim.x + threadIdx.x;
  if (t >= NN * MM) return;
  const int n = t / MM, m = t % MM;
  const float Bf = (float)BB;
  const float bu = beta_u[m], ba = beta_a[m];

  // init activation: sigma = Sum_b (V-g)^2 + eps
  float costi = 0.0f;
  for (int d = 0; d < PSZ; ++d) {
    float s1 = S1[(n * MM + m) * PSZ + d];
    float s2 = S2[(n * MM + m) * PSZ + d];
    float gv = g[((size_t)n * MM + m) * PSZ + d];
    float sig = s2 - 2.0f * gv * s1 + Bf * gv * gv + EPSF;
    costi += bu + 0.5f * logf(sig);
  }
  const float a_in = 1.0f / (1.0f + expf(-(LAM_INIT * (ba - costi))));

  // iteration 0: r uniform over b -> closed form from moments
  const float rm    = a_in / (float)MM;
  const float rsum  = Bf * rm;
  const float denom = rsum + EPSF;
  const float coeff = rm / denom;
  float costh = 0.0f;
  for (int d = 0; d < PSZ; ++d) {
    float s1 = S1[(n * MM + m) * PSZ + d];
    float s2 = S2[(n * MM + m) * PSZ + d];
    float mu = coeff * s1;
    float sg = coeff * (s2 - 2.0f * mu * s1 + Bf * mu * mu) + EPSF;
    MU[(n * MM + m) * PSZ + d] = mu;
    SG[(n * MM + m) * PSZ + d] = sg;
    costh += (bu + 0.5f * logf(sg)) * rsum;
  }
  AO[n * MM + m] = 1.0f / (1.0f + expf(-(LAM_ROUT * (ba - costh))));
}

// ---------------------------------------------------------------------------
// Kernel 4: EM iteration accumulation (iters 1,2).  16 lanes share one b
// (one lane per d); Sum_d ln p via shfl_xor within the 16-lane group; softmax
// over M in registers; moments Sum r, Sum rV, Sum rV^2 accumulated per thread
// then flushed with global_atomic_add_f32.
// ---------------------------------------------------------------------------
__global__ __launch_bounds__(256) void k_em_acc(
    const float* __restrict__ V,
    const float* __restrict__ MU, const float* __restrict__ SG,
    const float* __restrict__ AO,
    float* __restrict__ S0, float* __restrict__ S1, float* __restrict__ S2)
{
  __shared__ float mu_s[MM * PSZ], sg_s[MM * PSZ], a_s[MM], lna_s[MM];
  const int n = blockIdx.x >> 3;
  const int chunk = blockIdx.x & 7;
  for (int i = threadIdx.x; i < MM * PSZ; i += 256) {
    mu_s[i] = MU[n * MM * PSZ + i];
    sg_s[i] = SG[n * MM * PSZ + i];
  }
  if (threadIdx.x < MM) {
    float a = AO[n * MM + threadIdx.x];
    a_s[threadIdx.x]   = a;
    lna_s[threadIdx.x] = logf(a);
  }
  __syncthreads();

  const int d  = threadIdx.x & 15;
  const int bg = threadIdx.x >> 4;
  float s0p[MM] = {}, s1p[MM] = {}, s2p[MM] = {};

  for (int i = 0; i < 49; ++i) {
    const int b = chunk * 784 + bg + i * 16;
    const float* vr = V + ((size_t)n * BB + b) * MM * PSZ + d;
    float vm[MM], lnap[MM];
    float mx = -1e30f;
#pragma unroll
    for (int m = 0; m < MM; ++m) {
      float v = vr[m * PSZ];
      vm[m] = v;
      float dm = v - mu_s[m * PSZ + d];
      float sg = sg_s[m * PSZ + d];
      float lp = -(dm * dm) / (2.0f * sg) - 0.5f * logf(sg) - 0.5f * LN2PI;
      lp += __shfl_xor(lp, 1, 16);
      lp += __shfl_xor(lp, 2, 16);
      lp += __shfl_xor(lp, 4, 16);
      lp += __shfl_xor(lp, 8, 16);
      lp += lna_s[m];
      lnap[m] = lp;
      mx = fmaxf(mx, lp);
    }
    float se = 0.0f;
#pragma unroll
    for (int m = 0; m < MM; ++m) { lnap[m] = expf(lnap[m] - mx); se += lnap[m]; }
    const float inv = 1.0f / se;
#pragma unroll
    for (int m = 0; m < MM; ++m) {
      float r  = lnap[m] * inv * a_s[m];
      float rv = r * vm[m];
      s0p[m] += r;
      s1p[m] += rv;
      s2p[m] += rv * vm[m];
    }
  }
#pragma unroll
  for (int m = 0; m < MM; ++m) {
    if (d == 0) atomicAdd(&S0[n * MM + m], s0p[m]);
    atomicAdd(&S1[(n * MM + m) * PSZ + d], s1p[m]);
    atomicAdd(&S2[(n * MM + m) * PSZ + d], s2p[m]);
  }
}

// ---------------------------------------------------------------------------
// Kernel 5: finalize EM iteration from moments; last iter writes d_out.
// ---------------------------------------------------------------------------
__global__ void k_finalize(
    const float* __restrict__ S0, const float* __restrict__ S1,
    const float* __restrict__ S2, const float* __restrict__ beta_a,
    const float* __restrict__ beta_u,
    float* __restrict__ MU, float* __restrict__ SG, float* __restrict__ AO,
    int write_out, float* __restrict__ out_a, float* __restrict__ out_mu)
{
  int t = blockIdx.x * blockDim.x + threadIdx.x;
  if (t >= NN * MM) return;
  const int n = t / MM, m = t % MM;
  const float bu = beta_u[m], ba = beta_a[m];
  const float rsum  = S0[n * MM + m];
  const float denom = rsum + EPSF;
  float costh = 0.0f;
  for (int d = 0; d < PSZ; ++d) {
    float s1 = S1[(n * MM + m) * PSZ + d];
    float s2 = S2[(n * MM + m) * PSZ + d];
    float mu = s1 / denom;
    // Sum coeff (v-mu)^2 = (S2 - 2 mu S1 + mu^2 S0) / denom
    float sg = (s2 - 2.0f * mu * s1 + mu * mu * rsum) / denom + EPSF;
    MU[(n * MM + m) * PSZ + d] = mu;
    SG[(n * MM + m) * PSZ + d] = sg;
    costh += (bu + 0.5f * logf(sg)) * rsum;
    if (write_out) out_mu[(n * MM + m) * PSZ + d] = mu;
  }
  float a = 1.0f / (1.0f + expf(-(LAM_ROUT * (ba - costh))));
  AO[n * MM + m] = a;
  if (write_out) out_a[n * MM + m] = a;
}

// ---------------------------------------------------------------------------
extern "C" void kernel_launch(void* const* d_in, const int* in_sizes, int n_in,
                              void* d_out, int out_size, void* d_ws,
                              size_t ws_size, hipStream_t stream)
{
  const float* l      = (const float*)d_in[0];   // (32,32,16,14,14)
  const float* g      = (const float*)d_in[1];   // (32,10,16)
  const float* w      = (const float*)d_in[2];   // (1,32,1,10,4,4)
  const float* beta_a = (const float*)d_in[3];   // (10,1)
  const float* beta_u = (const float*)d_in[4];   // (10,1)
  float* out = (float*)d_out;                    // 320 a + 5120 mu

  // workspace layout (floats)
  const size_t NV = (size_t)NN * BB * MM * PSZ;  // votes: 32,112,640 floats
  float* ws = (float*)d_ws;
  float* V  = ws;
  float* S0 = V + NV;                 // 320
  float* S1 = S0 + NN * MM;           // 5120
  float* S2 = S1 + NN * MM * PSZ;     // 5120
  float* MU = S2 + NN * MM * PSZ;     // 5120
  float* SG = MU + NN * MM * PSZ;     // 5120
  float* AO = SG + NN * MM * PSZ;     // 320

  // 1) votes via WMMA: 32c * 32n * 49 row-tiles = 50176 waves, 3 WMMAs each
  k_votes_wmma<<<6272, 256, 0, stream>>>(l, w, V);

  // 2) init + iteration-0 moments
  const int nS12 = 2 * NN * MM * PSZ;                 // S1,S2 contiguous
  k_zero<<<(nS12 + 255) / 256, 256, 0, stream>>>(S1, nS12);
  k_init_acc<<<NN * 8, 256, 0, stream>>>(V, S1, S2);
  k_finalize0<<<(NN * MM + 255) / 256, 256, 0, stream>>>(S1, S2, g, beta_a,
                                                         beta_u, MU, SG, AO);

  // 3) EM iterations 1..2
  const int nS012 = NN * MM + 2 * NN * MM * PSZ;      // S0,S1,S2 contiguous
  for (int it = 1; it < 3; ++it) {
    k_zero<<<(nS012 + 255) / 256, 256, 0, stream>>>(S0, nS012);
    k_em_acc<<<NN * 8, 256, 0, stream>>>(V, MU, SG, AO, S0, S1, S2);
    k_finalize<<<(NN * MM + 255) / 256, 256, 0, stream>>>(
        S0, S1, S2, beta_a, beta_u, MU, SG, AO, (it == 2) ? 1 : 0, out,
        out + NN * MM);
  }
}